// SamplingNormalization_3813930959350
// MI455X (gfx1250) — compile-verified
//
#include <hip/hip_runtime.h>
#include <hip/hip_bf16.h>

typedef __attribute__((ext_vector_type(2))) float v2f;
typedef __attribute__((ext_vector_type(8))) float v8f;

#define N_PTS 16384
#define TILE 16
#define NTILES (N_PTS / TILE)   // 1024

// One wave32 per 16-row strip. The Gaussian kernel argument (in log2 domain)
// is produced directly by V_WMMA_F32_16X16X4_F32:
//   A_i = [2c*x_i, 2c*y_i, -c*sq_i, -c],  B_j = [x_j, y_j, 1, sq_j]
//   => D[i][j] = -(log2 e)/(2 sigma^2) * ||p_i - p_j||^2
// so kernel value = exp2(D[i][j]).
//
// batch is sorted, so the column tiles sharing a batch with this strip form a
// contiguous range found by binary search; no per-tile test needed inside the
// loop, and the loop bounds are wave-uniform scalars (full EXEC for WMMA).
__global__ __launch_bounds__(256)
void rbf_denom_kernel(const float* __restrict__ pos,
                      const float* __restrict__ weights,
                      const long long* __restrict__ batch,
                      const float* __restrict__ sigmaPtr,
                      float* __restrict__ out) {
  const int lane  = threadIdx.x & 31;
  const int wave  = threadIdx.x >> 5;
  const int strip = blockIdx.x * 8 + wave;   // which 16-row strip
  const int half  = lane >> 4;               // 0: lanes 0-15, 1: lanes 16-31
  const int c16   = lane & 15;

  const float sigma = sigmaPtr[0];
  const float c = 0.7213475204444817f / (sigma * sigma); // log2(e) / (2*sigma^2)

  // ---- A matrix for this strip.
  // ISA 32-bit 16x4 A layout: VGPR0 = K0 (lanes 0-15) / K2 (lanes 16-31),
  //                           VGPR1 = K1 / K3 ; M = lane%16 in both halves.
  const int   row = strip * TILE + c16;
  const float xr  = pos[2 * row];
  const float yr  = pos[2 * row + 1];
  const float sqr = xr * xr + yr * yr;
  v2f a;
  a.x = half ? (-c * sqr) : (2.0f * c * xr);  // K=2 : K=0
  a.y = half ? (-c)       : (2.0f * c * yr);  // K=3 : K=1

  // Batch ids of the 8 rows this lane's accumulator slots correspond to
  // (D layout: slot v, half h -> row = v + 8h).
  int rb[8];
#pragma unroll
  for (int v = 0; v < 8; ++v)
    rb[v] = (int)batch[strip * TILE + half * 8 + v];

  const int rowLo = (int)batch[strip * TILE];
  const int rowHi = (int)batch[strip * TILE + TILE - 1];

  // Binary search the contiguous same-batch column range [jStart, jEnd).
  int lo = 0, hi = N_PTS;
  while (lo < hi) {                     // lower_bound: first j with batch[j] >= rowLo
    int mid = (lo + hi) >> 1;
    if ((int)batch[mid] < rowLo) lo = mid + 1; else hi = mid;
  }
  const int jStart = lo;
  lo = 0; hi = N_PTS;
  while (lo < hi) {                     // upper_bound: first j with batch[j] > rowHi
    int mid = (lo + hi) >> 1;
    if ((int)batch[mid] <= rowHi) lo = mid + 1; else hi = mid;
  }
  const int jEnd = lo;

  // Tile bounds, forced wave-uniform (scalar loop control, EXEC stays all-1s).
  const int tStart = __builtin_amdgcn_readfirstlane(jStart >> 4);
  const int tEnd   = __builtin_amdgcn_readfirstlane((jEnd + TILE - 1) >> 4);

  float sums[8];
#pragma unroll
  for (int v = 0; v < 8; ++v) sums[v] = 0.0f;

  const v8f czero = {};

  for (int t = tStart; t < tEnd; ++t) {
    const int   col = t * TILE + c16;
    const float xc  = pos[2 * col];
    const float yc  = pos[2 * col + 1];
    const float sqc = xc * xc + yc * yc;
    const int   cb  = (int)batch[col];

    // B matrix 4x16, same K striping as A: VGPR0 = K0/K2, VGPR1 = K1/K3.
    v2f b;
    b.x = half ? 1.0f : xc;   // K=2 : K=0
    b.y = half ? sqc  : yc;   // K=3 : K=1

    v8f d = __builtin_amdgcn_wmma_f32_16x16x4_f32(
        false, a, false, b, (short)0, czero, false, false);

#pragma unroll
    for (int v = 0; v < 8; ++v) {
      float e = __builtin_amdgcn_exp2f(d[v]);          // v_exp_f32
      sums[v] += (rb[v] == cb) ? e : 0.0f;             // batch mask (boundary tiles)
    }
  }

  // Reduce across the 16 lanes of each half (the 16 columns of the tile row).
#pragma unroll
  for (int v = 0; v < 8; ++v) {
    float s = sums[v];
    s += __shfl_xor(s, 1, 32);
    s += __shfl_xor(s, 2, 32);
    s += __shfl_xor(s, 4, 32);
    s += __shfl_xor(s, 8, 32);
    sums[v] = s;
  }

  // Lane 0 of each half writes its 8 rows: out = weight / denom.
  if (c16 == 0) {
#pragma unroll
    for (int v = 0; v < 8; ++v) {
      int orow = strip * TILE + half * 8 + v;
      out[orow] = weights[orow] / sums[v];
    }
  }
}

extern "C" void kernel_launch(void* const* d_in, const int* in_sizes, int n_in,
                              void* d_out, int out_size, void* d_ws, size_t ws_size,
                              hipStream_t stream) {
  const float*      pos   = (const float*)d_in[0];      // [N,2] f32
  const float*      w     = (const float*)d_in[1];      // [N]   f32
  const long long*  batch = (const long long*)d_in[2];  // [N]   i64 (sorted)
  const float*      sigma = (const float*)d_in[3];      // [1]   f32
  float*            out   = (float*)d_out;              // [N]   f32
  (void)d_ws; (void)ws_size; (void)in_sizes; (void)n_in; (void)out_size;

  rbf_denom_kernel<<<NTILES / 8, 256, 0, stream>>>(pos, w, batch, sigma, out);
}